// DecoderLayer_73735998538231
// MI455X (gfx1250) — compile-verified
//
#include <hip/hip_runtime.h>
#include <hip/hip_bf16.h>

// ---------------------------------------------------------------------------
// DecoderLayer on MI455X (gfx1250, wave32)
//
// Strategy: conv-as-GEMM-over-channels with all 9 taps in the WMMA N dim:
//   Z[p, t] = sum_ch X[ch, p] * W[ch, t]        (V_WMMA_F32_16X16X4_F32, K=16 via 4 steps)
//   y[r,c]  = sum_{dy,dx} Z[(r+dy-1, c+dx-1), 3*dy+dx]   (LDS shift-add)
// Full-precision f32 WMMA; Z kept entirely in the 320KB WGP LDS (144KB).
// One workgroup (8 wave32) per node; 1024 WMMA per node.
// ---------------------------------------------------------------------------

#define NODES   1024
#define FANIN   16
#define NSP     64
#define PIX     (NSP * NSP)     // 4096
#define THRESH  12
#define WG      256             // 8 waves of 32
#define NTILES  (PIX / 16)      // 256 pixel tiles of 16

typedef __attribute__((ext_vector_type(2))) float v2f;
typedef __attribute__((ext_vector_type(8))) float v8f;

__launch_bounds__(WG, 2)
__global__ void decoder_wmma_kernel(const float* __restrict__ prev,          // (1024,64,64) f32
                                    const unsigned char* __restrict__ act,   // (1024,) bool
                                    const int* __restrict__ pidx,            // (1024,16) i32
                                    const float* __restrict__ W,             // (1024,16,3,3) f32
                                    const float* __restrict__ bias,          // (1024,) f32
                                    float* __restrict__ out,                 // (1024,64,64) f32
                                    float* __restrict__ out_act)             // (1024,) f32 0/1
{
    extern __shared__ float Zs[];      // PIX * 9 floats = 147456 B (dynamic LDS)
    __shared__ int s_parent[FANIN];
    __shared__ int s_active;

    const int k   = blockIdx.x;
    const int tid = threadIdx.x;

    // ---- gate + parent index staging (workgroup-uniform result) ----
    if (tid < FANIN) {
        s_parent[tid] = pidx[k * FANIN + tid];
    }
    if (tid == 0) {
        int cnt = 0;
        for (int i = 0; i < FANIN; ++i)
            cnt += (act[pidx[k * FANIN + i]] != 0) ? 1 : 0;
        s_active = (cnt >= THRESH) ? 1 : 0;
        out_act[k] = (float)s_active;
    }
    __syncthreads();

    if (!s_active) {
        // Inactive node: emit zero tensor (uniform branch -> no EXEC hazard).
        for (int i = tid; i < PIX; i += WG)
            out[(size_t)k * PIX + i] = 0.0f;
        return;
    }

    const int lane = tid & 31;
    const int n    = lane & 15;   // A row (pixel-in-tile) AND B column (tap id)
    const int h    = lane >> 4;   // K-half: VGPR0 holds K=2h, VGPR1 holds K=2h+1
    const int wave = tid >> 5;    // 0..7

    // ---- loop-invariant B operand: B[k_ch][tap], taps 9..15 zero-padded ----
    // B 4x16 layout (mirror of A): VGPR j, lane L -> k = 2*(L>>4) + j, n = L&15
    const float* Wk = W + (size_t)k * FANIN * 9;
    v2f bop[4];
#pragma unroll
    for (int s = 0; s < 4; ++s) {
        const int c0 = 4 * s + 2 * h;
        bop[s].x = (n < 9) ? Wk[(c0 + 0) * 9 + n] : 0.0f;
        bop[s].y = (n < 9) ? Wk[(c0 + 1) * 9 + n] : 0.0f;
    }

    // ---- per-lane channel base pointers (channels this lane feeds into A) ----
    const float* xbase[8];
#pragma unroll
    for (int s = 0; s < 4; ++s) {
        xbase[2 * s + 0] = prev + (size_t)s_parent[4 * s + 2 * h + 0] * PIX;
        xbase[2 * s + 1] = prev + (size_t)s_parent[4 * s + 2 * h + 1] * PIX;
    }

    // ---- Phase 1: Z[p, t] tiles via V_WMMA_F32_16X16X4_F32 ----
    for (int T = wave; T < NTILES; T += (WG / 32)) {
        const int p0 = T * 16;

        // prefetch next tile for this wave (global_prefetch_b8 path)
        if (T + (WG / 32) < NTILES) {
            __builtin_prefetch(xbase[0] + p0 + 16 * (WG / 32) + n, 0, 1);
            __builtin_prefetch(xbase[4] + p0 + 16 * (WG / 32) + n, 0, 1);
        }

        v8f c = {};
#pragma unroll
        for (int s = 0; s < 4; ++s) {
            // A 16x4 layout: lane L -> pixel p0 + (L&15); VGPR0: ch=4s+2h, VGPR1: ch=4s+2h+1
            v2f a;
            a.x = xbase[2 * s + 0][p0 + n];
            a.y = xbase[2 * s + 1][p0 + n];
            c = __builtin_amdgcn_wmma_f32_16x16x4_f32(
                    /*neg_a=*/false, a, /*neg_b=*/false, bop[s],
                    /*c_mod=*/(short)0, c, /*reuse_a=*/false, /*reuse_b=*/false);
        }

        // C/D 16x16 layout: lane L, VGPR v -> row M = v + 8*(L>>4), col N = L&15.
        // Column n is this lane's tap; only taps 0..8 are meaningful.
        if (n < 9) {
            const int prow = p0 + 8 * h;
#pragma unroll
            for (int v = 0; v < 8; ++v)
                Zs[(size_t)(prow + v) * 9 + n] = c[v];
        }
    }
    __syncthreads();

    // ---- Phase 2: 9-tap shift-add from LDS, bias, tanh, store ----
    const float bk = bias[k];
    for (int q = tid; q < PIX; q += WG) {
        const int r  = q >> 6;
        const int cc = q & 63;
        float acc = bk;
#pragma unroll
        for (int dy = 0; dy < 3; ++dy) {
            const int pr = r + dy - 1;
            if ((unsigned)pr < (unsigned)NSP) {
#pragma unroll
                for (int dx = 0; dx < 3; ++dx) {
                    const int pc = cc + dx - 1;
                    if ((unsigned)pc < (unsigned)NSP)
                        acc += Zs[(size_t)(pr * NSP + pc) * 9 + (dy * 3 + dx)];
                }
            }
        }
        out[(size_t)k * PIX + q] = tanhf(acc);
    }
}

// ---------------------------------------------------------------------------
// Harness entry point
// Inputs (setup_inputs order):
//   0: prev_outputs   (1024,64,64) f32
//   1: prev_is_active (1024,)      bool (1 byte each)
//   2: parent_indices (1024,16)    i32
//   3: W              (1024,16,3,3) f32
//   4: b              (1024,)      f32
// Output: [ out (1024*4096) f32 | out_active (1024) f32(0/1) ]
// ---------------------------------------------------------------------------
extern "C" void kernel_launch(void* const* d_in, const int* in_sizes, int n_in,
                              void* d_out, int out_size, void* d_ws, size_t ws_size,
                              hipStream_t stream) {
    const float*         prev = (const float*)d_in[0];
    const unsigned char* act  = (const unsigned char*)d_in[1];
    const int*           pidx = (const int*)d_in[2];
    const float*         W    = (const float*)d_in[3];
    const float*         b    = (const float*)d_in[4];

    float* out     = (float*)d_out;
    float* out_act = out + (size_t)NODES * PIX;

    const size_t lds_bytes = (size_t)PIX * 9 * sizeof(float);  // 147456 B < 320 KB
    decoder_wmma_kernel<<<NODES, WG, lds_bytes, stream>>>(prev, act, pidx, W, b,
                                                          out, out_act);
}